// MultiHyperbolicGraphConvolution_54786602827841
// MI455X (gfx1250) — compile-verified
//
#include <hip/hip_runtime.h>
#include <stdint.h>

// MultiHyperbolicGraphConvolution for MI455X (gfx1250, wave32).
// No dense GEMM in the reference (scalar fusion weights only) -> WMMA has no
// legitimate role. CDNA5-specific paths used instead:
//   - global_load_async_to_lds_b128 + s_wait_asynccnt for edge streaming
//   - global_prefetch_b8 for the random row gathers
//   - native global_atomic_add_f32 scatter (node tables ~26MB, L2-resident in 192MB)
// Lane d-mapping is {2*lane, 2*lane+1} so every row access is one b64 per lane.

#define NU_ 60000
#define NI_ 40000
#define NT_ (NU_ + NI_)
#define D_  64
#define EPS_ 1e-10f
#define TCLIP_ (1.0f - 1e-5f)
#define TILE 256

// ---------------------------------------------------------------- zero (b128 stores)
__global__ void mhgc_zero(float4* __restrict__ p, int n4) {
    int i = blockIdx.x * blockDim.x + threadIdx.x;
    int stride = gridDim.x * blockDim.x;
    float4 z = {0.0f, 0.0f, 0.0f, 0.0f};
    for (; i < n4; i += stride) p[i] = z;
}

// ---------------------------------------------------------------- prep
// One wave32 per node: x -> expmap0(x,c); store lam*x_hyp and (lam-1).
__global__ void mhgc_prep(const float* __restrict__ x, float* __restrict__ lamx,
                          float* __restrict__ lamm1, int n_nodes, float c) {
    int node = (blockIdx.x * blockDim.x + threadIdx.x) >> 5;
    int lane = threadIdx.x & 31;
    if (node >= n_nodes) return;
    const float sc = sqrtf(c);
    size_t base = (size_t)node * D_ + 2 * lane;
    float2 xv = *(const float2*)&x[base];
    float ss = xv.x * xv.x + xv.y * xv.y;
    for (int o = 16; o; o >>= 1) ss += __shfl_xor(ss, o, 32);
    float nrm = fmaxf(sqrtf(ss), EPS_);
    float th  = tanhf(sc * nrm);
    float t   = th / (sc * nrm);        // expmap0 scale
    float xhn = th / sc;                // ||x_hyp||
    float lam = 2.0f / fmaxf(1.0f - c * xhn * xhn, EPS_);
    float2 o2 = {lam * t * xv.x, lam * t * xv.y};
    *(float2*)&lamx[base] = o2;
    if (lane == 0) lamm1[node] = lam - 1.0f;
}

// ---------------------------------------------------------------- scatter
// Stage 256-edge tiles of (rows, cols, vals) into LDS via the CDNA5 async
// copy engine (b128), then one wave per edge: b64 gather of lamx[col],
// fp32-atomic-add scatter into nom[row], lane0 adds den[row].
__global__ void mhgc_scatter(const int* __restrict__ rows, const int* __restrict__ cols,
                             const float* __restrict__ vals, int n_edges,
                             const float* __restrict__ lamx, const float* __restrict__ lamm1,
                             float* __restrict__ nom, float* __restrict__ den) {
    __shared__ __align__(16) int   s_r[TILE];
    __shared__ __align__(16) int   s_c[TILE];
    __shared__ __align__(16) float s_v[TILE];
    const int t    = threadIdx.x;
    const int lane = t & 31;
    const int wv   = t >> 5;

    for (int base = blockIdx.x * TILE; base < n_edges; base += gridDim.x * TILE) {
        int cnt = min(TILE, n_edges - base);
        if (cnt == TILE) {
            // async global -> LDS staging, 16B per thread, waves 0..5 only
            if (t < 192) {
                const int which = t >> 6;       // 0=rows, 1=cols, 2=vals
                const int idx   = (t & 63) * 4; // element offset inside tile
                uint64_t g;
                uint32_t l;
                if (which == 0) {
                    g = (uint64_t)(uintptr_t)(rows + base + idx);
                    l = (uint32_t)(uintptr_t)&s_r[idx];
                } else if (which == 1) {
                    g = (uint64_t)(uintptr_t)(cols + base + idx);
                    l = (uint32_t)(uintptr_t)&s_c[idx];
                } else {
                    g = (uint64_t)(uintptr_t)(vals + base + idx);
                    l = (uint32_t)(uintptr_t)&s_v[idx];
                }
                asm volatile("global_load_async_to_lds_b128 %0, %1, off"
                             :: "v"(l), "v"(g) : "memory");
            }
            asm volatile("s_wait_asynccnt 0" ::: "memory");
        } else {
            if (t < cnt) {
                s_r[t] = rows[base + t];
                s_c[t] = cols[base + t];
                s_v[t] = vals[base + t];
            }
        }
        __syncthreads();

        for (int k = wv; k < cnt; k += 8) {  // 8 waves per block
            int   r  = s_r[k];
            int   ci = s_c[k];
            float v  = s_v[k];
            if (k + 8 < cnt) {
                // emits global_prefetch_b8 for the next gather row
                __builtin_prefetch(&lamx[(size_t)s_c[k + 8] * D_ + 2 * lane], 0, 1);
            }
            size_t cb = (size_t)ci * D_ + 2 * lane;
            size_t rb = (size_t)r  * D_ + 2 * lane;
            float2 f = *(const float2*)&lamx[cb];   // one global_load_b64 per lane
            unsafeAtomicAdd(&nom[rb],     v * f.x);
            unsafeAtomicAdd(&nom[rb + 1], v * f.y);
            if (lane == 0) unsafeAtomicAdd(&den[r], v * lamm1[ci]);
        }
        __syncthreads();
    }
}

// ---------------------------------------------------------------- finalize
// One wave32 per node: two_mean = nom/clip(den); mobius_scalar_mul(0.5);
// logmap0; weight; write/accumulate into newh and accumulate into out.
__global__ void mhgc_finalize(const float* __restrict__ nom, const float* __restrict__ den,
                              float* __restrict__ newh, float* __restrict__ out,
                              int n_nodes, float c, float wA, float wB, int split,
                              int accumulate) {
    int node = (blockIdx.x * blockDim.x + threadIdx.x) >> 5;
    int lane = threadIdx.x & 31;
    if (node >= n_nodes) return;
    const float sc = sqrtf(c);
    size_t base = (size_t)node * D_ + 2 * lane;
    float d  = fmaxf(den[node], EPS_);
    float2 nv = *(const float2*)&nom[base];
    float t0 = nv.x / d;
    float t1 = nv.y / d;
    float ss = t0 * t0 + t1 * t1;
    for (int o = 16; o; o >>= 1) ss += __shfl_xor(ss, o, 32);
    float nrm = fmaxf(sqrtf(ss), EPS_);
    // mobius_scalar_mul(0.5, t, c)
    float a   = fminf(sc * nrm, TCLIP_);
    float ath = 0.5f * logf((1.0f + a) / (1.0f - a));   // atanh(a)
    float mth = tanhf(0.5f * ath);
    float ms  = mth / (sc * nrm);
    float m0  = ms * t0;
    float m1  = ms * t1;
    // logmap0(m, c); ||m|| = mth/sc
    float mn  = fmaxf(mth / sc, EPS_);
    float b   = fminf(sc * mn, TCLIP_);
    float ys  = (0.5f * logf((1.0f + b) / (1.0f - b))) / (sc * mn);
    float w   = (node < split) ? wA : wB;
    float y0  = w * ys * m0;
    float y1  = w * ys * m1;
    float2 yv = {y0, y1};
    if (accumulate) {
        float2 ph = *(float2*)&newh[base];
        yv.x += ph.x; yv.y += ph.y;
        *(float2*)&newh[base] = yv;
        float2 po = *(float2*)&out[base];
        po.x += y0; po.y += y1;
        *(float2*)&out[base] = po;
    } else {
        *(float2*)&newh[base] = yv;
        float2 po = *(float2*)&out[base];
        po.x += y0; po.y += y1;
        *(float2*)&out[base] = po;
    }
}

// ---------------------------------------------------------------- launch
extern "C" void kernel_launch(void* const* d_in, const int* in_sizes, int n_in,
                              void* d_out, int out_size, void* d_ws, size_t ws_size,
                              hipStream_t stream) {
    const float* xu      = (const float*)d_in[0];
    const float* xi      = (const float*)d_in[1];
    const int*   ui_rows = (const int*)d_in[2];
    const int*   ui_cols = (const int*)d_in[3];
    const float* ui_vals = (const float*)d_in[4];
    const int*   ii_rows = (const int*)d_in[5];
    const int*   ii_cols = (const int*)d_in[6];
    const float* ii_vals = (const float*)d_in[7];
    const int E_UI = in_sizes[2];
    const int E_II = in_sizes[5];

    float* ws    = (float*)d_ws;
    float* h     = ws;                 // NT_*D_
    float* nh    = h    + (size_t)NT_ * D_;
    float* lamx  = nh   + (size_t)NT_ * D_;
    float* nom   = lamx + (size_t)NT_ * D_;
    float* lamm1 = nom  + (size_t)NT_ * D_;
    float* den   = lamm1 + NT_;
    float* out   = (float*)d_out;      // NT_*D_

    hipMemcpyAsync(h, xu, (size_t)NU_ * D_ * sizeof(float),
                   hipMemcpyDeviceToDevice, stream);
    hipMemcpyAsync(h + (size_t)NU_ * D_, xi, (size_t)NI_ * D_ * sizeof(float),
                   hipMemcpyDeviceToDevice, stream);

    const int ZB = 256;
    mhgc_zero<<<1024, ZB, 0, stream>>>((float4*)out, NT_ * D_ / 4);

    float* cur = h;
    float* nxt = nh;
    const float c_ui = 1.0f, c_ii = 0.5f;
    const float W_USER_UI = 1.0f, W_ITEM_UI = 0.6f, W_ITEM_II = 0.4f;

    auto nodeBlocks = [](int n) { return (n + 7) / 8; };  // 8 waves (nodes) / 256-thr block
    auto edgeBlocks = [](int e) {
        int nb = (e + TILE - 1) / TILE;
        return nb > 16384 ? 16384 : nb;
    };

    for (int layer = 0; layer < 2; ++layer) {
        // ----- space "ui": all NT_ nodes, bipartite edges -----
        mhgc_zero<<<1024, ZB, 0, stream>>>((float4*)nom, NT_ * D_ / 4);
        mhgc_zero<<<128,  ZB, 0, stream>>>((float4*)den, NT_ / 4);
        mhgc_prep<<<nodeBlocks(NT_), 256, 0, stream>>>(cur, lamx, lamm1, NT_, c_ui);
        mhgc_scatter<<<edgeBlocks(E_UI), 256, 0, stream>>>(ui_rows, ui_cols, ui_vals, E_UI,
                                                           lamx, lamm1, nom, den);
        mhgc_finalize<<<nodeBlocks(NT_), 256, 0, stream>>>(nom, den, nxt, out, NT_, c_ui,
                                                           W_USER_UI, W_ITEM_UI, NU_, 0);

        // ----- space "ii": NI_ item nodes -----
        mhgc_zero<<<1024, ZB, 0, stream>>>((float4*)nom, NI_ * D_ / 4);
        mhgc_zero<<<128,  ZB, 0, stream>>>((float4*)den, NI_ / 4);
        mhgc_prep<<<nodeBlocks(NI_), 256, 0, stream>>>(cur + (size_t)NU_ * D_, lamx, lamm1,
                                                       NI_, c_ii);
        mhgc_scatter<<<edgeBlocks(E_II), 256, 0, stream>>>(ii_rows, ii_cols, ii_vals, E_II,
                                                           lamx, lamm1, nom, den);
        mhgc_finalize<<<nodeBlocks(NI_), 256, 0, stream>>>(nom, den,
                                                           nxt + (size_t)NU_ * D_,
                                                           out + (size_t)NU_ * D_,
                                                           NI_, c_ii,
                                                           W_ITEM_II, W_ITEM_II, 0, 1);
        // swap h / newh for next layer
        float* tmp = cur; cur = nxt; nxt = tmp;
    }
}